// TripletLoss_79534204387964
// MI455X (gfx1250) — compile-verified
//
#include <hip/hip_runtime.h>

#define NN 384
#define DD 128
#define MARGIN 0.3f

typedef __attribute__((ext_vector_type(2))) float v2f;
typedef __attribute__((ext_vector_type(8))) float v8f;

// ---------------------------------------------------------------------------
// sq[i] = sum_k E[i,k]^2   — one wave32 per row, float4 coalesced load.
// ---------------------------------------------------------------------------
__global__ __launch_bounds__(32) void sq_kernel(const float* __restrict__ E,
                                                float* __restrict__ sq) {
  const int row = blockIdx.x;
  const int lane = threadIdx.x;  // 0..31; 32 lanes * float4 = 128 elements
  const float4 v = ((const float4*)(E + row * DD))[lane];
  float s = v.x * v.x + v.y * v.y + v.z * v.z + v.w * v.w;
#pragma unroll
  for (int off = 16; off > 0; off >>= 1) s += __shfl_xor(s, off, 32);
  if (lane == 0) sq[row] = s;
}

// ---------------------------------------------------------------------------
// dist[i,j] = sq[i] + sq[j] - 2 * (E E^T)[i,j]
// One wave per 16x16 tile; K-loop of 32 x V_WMMA_F32_16X16X4_F32.
// A 16x4 f32 layout (ISA 7.12.2): lanes 0-15 hold M=lane, K={k,k+1} in v0/v1;
// lanes 16-31 hold M=lane-16, K={k+2,k+3}. B mirrors with N in the lane index.
// C/D: lane<16 -> (M=v, N=lane); lane>=16 -> (M=v+8, N=lane-16).
// ---------------------------------------------------------------------------
__global__ __launch_bounds__(32) void dist_wmma(const float* __restrict__ E,
                                                const float* __restrict__ sq,
                                                float* __restrict__ dist) {
  const int i0 = blockIdx.y * 16;
  const int j0 = blockIdx.x * 16;
  const int lane = threadIdx.x;
  const int lane_lo = lane & 15;
  const int sel = lane >> 4;  // 0 for lanes 0-15, 1 for lanes 16-31

  const float* __restrict__ arow = E + (i0 + lane_lo) * DD;
  const float* __restrict__ brow = E + (j0 + lane_lo) * DD;

  v8f acc = {};
#pragma unroll 8
  for (int k = 0; k < DD; k += 4) {
    const int ka = k + 2 * sel;  // even -> 8B aligned
    v2f a = *(const v2f*)(arow + ka);
    v2f b = *(const v2f*)(brow + ka);
    // 8 args: (neg_a, A, neg_b, B, c_mod, C, reuse_a, reuse_b)
    acc = __builtin_amdgcn_wmma_f32_16x16x4_f32(
        false, a, false, b, (short)0, acc, false, false);
  }

  const float sqj = sq[j0 + lane_lo];
#pragma unroll
  for (int v = 0; v < 8; ++v) {
    const int m = v + 8 * sel;
    const float d = sq[i0 + m] + sqj - 2.0f * acc[v];
    dist[(i0 + m) * NN + (j0 + lane_lo)] = d;
  }
}

// ---------------------------------------------------------------------------
// Zero the two output scalars (d_out is poisoned by the harness).
// ---------------------------------------------------------------------------
__global__ void zero_out(float* out) {
  if (threadIdx.x == 0) {
    out[0] = 0.0f;
    out[1] = 0.0f;
  }
}

// ---------------------------------------------------------------------------
// Per-anchor reduction. Block i: stage dist row i + labels in LDS, compact
// positive distances (same label, incl. self) into a list, then each thread
// scans negatives against the broadcast positive list:
//   L = dp - dn + margin, keep if 0 < L < margin; sum L and count.
// ---------------------------------------------------------------------------
__global__ __launch_bounds__(256) void reduce_kernel(const float* __restrict__ dist,
                                                     const int* __restrict__ labels,
                                                     float* __restrict__ out) {
  __shared__ float s_row[NN];
  __shared__ float s_pos[NN];
  __shared__ int s_lab[NN];
  __shared__ int s_npos;
  __shared__ float s_red[256];

  const int i = blockIdx.x;
  const int tid = threadIdx.x;

  if (tid == 0) s_npos = 0;
  for (int j = tid; j < NN; j += 256) {
    s_row[j] = dist[i * NN + j];
    s_lab[j] = labels[j];
  }
  __syncthreads();

  const int labI = s_lab[i];

  // Compact positives (order irrelevant for the sum).
  for (int j = tid; j < NN; j += 256) {
    if (s_lab[j] == labI) {
      const int idx = atomicAdd(&s_npos, 1);
      s_pos[idx] = s_row[j];
    }
  }
  __syncthreads();

  const int npos = s_npos;
  float loss = 0.0f;
  float cnt = 0.0f;

  for (int n = tid; n < NN; n += 256) {
    if (s_lab[n] != labI) {
      const float dn = s_row[n];
      for (int q = 0; q < npos; ++q) {        // LDS broadcast read
        const float L = s_pos[q] - dn + MARGIN;
        if (L > 0.0f && L < MARGIN) {
          loss += L;
          cnt += 1.0f;
        }
      }
    }
  }

  // Block tree reduction: loss then count.
  s_red[tid] = loss;
  __syncthreads();
#pragma unroll
  for (int s = 128; s > 0; s >>= 1) {
    if (tid < s) s_red[tid] += s_red[tid + s];
    __syncthreads();
  }
  if (tid == 0) atomicAdd(&out[0], s_red[0]);
  __syncthreads();

  s_red[tid] = cnt;
  __syncthreads();
#pragma unroll
  for (int s = 128; s > 0; s >>= 1) {
    if (tid < s) s_red[tid] += s_red[tid + s];
    __syncthreads();
  }
  if (tid == 0) atomicAdd(&out[1], s_red[0]);
}

// ---------------------------------------------------------------------------
extern "C" void kernel_launch(void* const* d_in, const int* in_sizes, int n_in,
                              void* d_out, int out_size, void* d_ws, size_t ws_size,
                              hipStream_t stream) {
  const float* E = (const float*)d_in[0];     // [384,128] f32
  const int* labels = (const int*)d_in[1];    // [384] i32
  float* out = (float*)d_out;                 // [2] f32: losses, count

  float* dist = (float*)d_ws;                 // 384*384 f32
  float* sq = dist + NN * NN;                 // 384 f32

  sq_kernel<<<dim3(NN), dim3(32), 0, stream>>>(E, sq);
  dist_wmma<<<dim3(NN / 16, NN / 16), dim3(32), 0, stream>>>(E, sq, dist);
  zero_out<<<1, 32, 0, stream>>>(out);
  reduce_kernel<<<dim3(NN), dim3(256), 0, stream>>>(dist, labels, out);
}